// SparseGCM_29197187678591
// MI455X (gfx1250) — compile-verified
//
#include <hip/hip_runtime.h>
#include <hip/hip_bf16.h>
#include <math.h>

#define BB   128
#define TT   16
#define FEAT 256
#define NN   1024
#define EE   8192

typedef float v2f __attribute__((ext_vector_type(2)));
typedef float v8f __attribute__((ext_vector_type(8)));

__device__ __forceinline__ float fast_tanh(float x) {
#if __has_builtin(__builtin_amdgcn_tanhf)
    return __builtin_amdgcn_tanhf(x);     // native V_TANH_F32 (trans op)
#else
    return tanhf(x);
#endif
}

// ---------------- prep kernels ----------------
__global__ void copy_nodes_k(const float4* __restrict__ src, float4* __restrict__ dst, size_t n4) {
    size_t i = (size_t)blockIdx.x * blockDim.x + threadIdx.x;
    if (i < n4) dst[i] = src[i];
}

// scatter x[b, r, :] into nodes[b, T[b]+r, :]
__global__ void scatter_x_k(const float4* __restrict__ x4, const int* __restrict__ T,
                            float4* __restrict__ nodes4) {
    size_t i = (size_t)blockIdx.x * blockDim.x + threadIdx.x;  // over B*t*FEAT/4
    if (i >= (size_t)BB * TT * (FEAT / 4)) return;
    int b   = (int)(i >> 10);            // t*FEAT/4 = 1024
    int rem = (int)(i & 1023);
    int row = rem >> 6;                  // FEAT/4 = 64
    int c4  = rem & 63;
    size_t drow = (size_t)b * NN + (size_t)(T[b] + row);
    nodes4[drow * (FEAT / 4) + c4] = x4[i];
}

// pass-through outputs: edges (as float), weights, T+taus
__global__ void misc_out_k(const int* __restrict__ edges, const float* __restrict__ weights,
                           const int* __restrict__ T, const int* __restrict__ taus,
                           float* __restrict__ oe, float* __restrict__ ow, float* __restrict__ ot) {
    size_t i = (size_t)blockIdx.x * blockDim.x + threadIdx.x;
    const size_t nE2 = (size_t)BB * 2 * EE;
    const size_t nW  = (size_t)BB * EE;
    if (i < nE2) {
        oe[i] = (float)edges[i];
    } else if (i < nE2 + nW) {
        ow[i - nE2] = weights[i - nE2];
    } else if (i < nE2 + nW + BB) {
        int b = (int)(i - nE2 - nW);
        ot[b] = (float)(T[b] + taus[b]);
    }
}

// transpose one 256x256 f32 matrix: D[n][k] = S[k][n]   (block 32x8, grid 8x8)
__global__ void transpose_k(const float* __restrict__ S, float* __restrict__ D) {
    __shared__ float tile[32][33];
    int tx = threadIdx.x;
    int bx = blockIdx.x * 32;            // n-range of D rows
    int by = blockIdx.y * 32;            // k-range of D cols
    for (int j = threadIdx.y; j < 32; j += 8)
        tile[j][tx] = S[(size_t)(by + j) * FEAT + bx + tx];
    __syncthreads();
    for (int j = threadIdx.y; j < 32; j += 8)
        D[(size_t)(bx + j) * FEAT + by + tx] = tile[tx][j];
}

// ---------------- edge aggregation: agg[b,dst] += w * h[b,src] ----------------
__global__ void agg_k(const float* __restrict__ h, const int* __restrict__ edges,
                      const float* __restrict__ weights, float* __restrict__ agg) {
    int gw   = (int)(((size_t)blockIdx.x * blockDim.x + threadIdx.x) >> 5);
    int lane = threadIdx.x & 31;
    if (gw >= BB * EE) return;
    int b = gw / EE;
    int e = gw - b * EE;
    const int* eb = edges + (size_t)b * 2 * EE;
    int s = eb[e];
    int d = eb[EE + e];
    if (s < 0 || d < 0) return;                    // invalid edge -> weight 0
    float w = weights[(size_t)b * EE + e];
    const float* hs = h + ((size_t)b * NN + (size_t)s) * FEAT;
    float* ad = agg + ((size_t)b * NN + (size_t)d) * FEAT;
#pragma unroll
    for (int j = 0; j < FEAT / 32; ++j) {
        int c = lane + 32 * j;
        unsafeAtomicAdd(&ad[c], w * hs[c]);        // global_atomic_add_f32
    }
}

// ---------------- fused dual-GEMM + tanh on one 16-row strip ----------------
// O[16,256] = tanh(Hrow @ Ws + Arow @ Wm), weights given TRANSPOSED (Wt[n][k])
// so each B fragment is one contiguous 8-byte (v2f) load.
__device__ __forceinline__ void tile16_gemm(const float* __restrict__ Hrow,
                                            const float* __restrict__ Arow,
                                            const float* __restrict__ Wst,
                                            const float* __restrict__ Wmt,
                                            float* __restrict__ Orow) {
    __shared__ __align__(16) float sH[16][FEAT + 4];
    __shared__ __align__(16) float sA[16][FEAT + 4];
    int tid = threadIdx.x;
    for (int i = tid; i < 16 * (FEAT / 4); i += 256) {
        int r = i >> 6, c4 = i & 63;
        *(float4*)&sH[r][c4 * 4] = ((const float4*)(Hrow + (size_t)r * FEAT))[c4];
        *(float4*)&sA[r][c4 * 4] = ((const float4*)(Arow + (size_t)r * FEAT))[c4];
    }
    __syncthreads();

    int wave = tid >> 5, lane = tid & 31;
    int lo = lane & 15, hi = lane >> 4;            // ISA 32-bit A/C fragment layout
#pragma unroll
    for (int tt = 0; tt < 2; ++tt) {
        int n0 = (wave * 2 + tt) * 16;
        v8f c = {};
        const float* wsrow = Wst + (size_t)(n0 + lo) * FEAT;  // row n of W^T = col n of W
        const float* wmrow = Wmt + (size_t)(n0 + lo) * FEAT;
        for (int k4 = 0; k4 < FEAT; k4 += 4) {
            int ka = k4 + 2 * hi;                  // lanes 16-31 hold K=2,3
            v2f a = *(const v2f*)&sH[lo][ka];
            v2f b = *(const v2f*)&wsrow[ka];       // global_load_b64
            c = __builtin_amdgcn_wmma_f32_16x16x4_f32(false, a, false, b,
                                                      (short)0, c, false, false);
            a = *(const v2f*)&sA[lo][ka];
            b = *(const v2f*)&wmrow[ka];
            c = __builtin_amdgcn_wmma_f32_16x16x4_f32(false, a, false, b,
                                                      (short)0, c, false, false);
        }
#pragma unroll
        for (int v = 0; v < 8; ++v) {
            int r = v + 8 * hi;                    // C VGPR v -> rows v / v+8
            Orow[(size_t)r * FEAT + n0 + lo] = fast_tanh(c[v]);
        }
    }
}

// layer 1: all B*N rows
__global__ void gemm_full_k(const float* __restrict__ H, const float* __restrict__ AGG,
                            const float* __restrict__ Wst, const float* __restrict__ Wmt,
                            float* __restrict__ O) {
    size_t row0 = (size_t)blockIdx.x * 16;
    tile16_gemm(H + row0 * FEAT, AGG + row0 * FEAT, Wst, Wmt, O + row0 * FEAT);
}

// layer 2: only the gathered rows T[b]..T[b]+15 per batch -> mx directly
__global__ void gemm_out_k(const float* __restrict__ H, const float* __restrict__ AGG,
                           const float* __restrict__ Wst, const float* __restrict__ Wmt,
                           const int* __restrict__ T, float* __restrict__ MX) {
    int b = blockIdx.x;
    size_t row0 = (size_t)b * NN + (size_t)T[b];
    tile16_gemm(H + row0 * FEAT, AGG + row0 * FEAT, Wst, Wmt, MX + (size_t)b * TT * FEAT);
}

extern "C" void kernel_launch(void* const* d_in, const int* in_sizes, int n_in,
                              void* d_out, int out_size, void* d_ws, size_t ws_size,
                              hipStream_t stream) {
    const float* x       = (const float*)d_in[0];
    const int*   taus    = (const int*)d_in[1];
    const float* nodes   = (const float*)d_in[2];
    const int*   edges   = (const int*)d_in[3];
    const float* weights = (const float*)d_in[4];
    const int*   T       = (const int*)d_in[5];
    const float* Ws1     = (const float*)d_in[6];
    const float* Wm1     = (const float*)d_in[7];
    const float* Ws2     = (const float*)d_in[8];
    const float* Wm2     = (const float*)d_in[9];

    float* out = (float*)d_out;
    const size_t nMX    = (size_t)BB * TT * FEAT;
    const size_t nNODES = (size_t)BB * NN * FEAT;
    const size_t nE2    = (size_t)BB * 2 * EE;
    const size_t nW     = (size_t)BB * EE;
    const size_t nWMAT  = (size_t)FEAT * FEAT;
    float* o_mx    = out;
    float* o_nodes = o_mx + nMX;
    float* o_edges = o_nodes + nNODES;
    float* o_w     = o_edges + nE2;
    float* o_t     = o_w + nW;

    float* agg  = (float*)d_ws;        // 128 MB
    float* h1   = agg + nNODES;        // 128 MB
    float* w1st = h1 + nNODES;         // 4 x 256 KB transposed weights
    float* w1mt = w1st + nWMAT;
    float* w2st = w1mt + nWMAT;
    float* w2mt = w2st + nWMAT;

    // weight transposes (tiny, reused by every GEMM block)
    dim3 tb(32, 8), tg(FEAT / 32, FEAT / 32);
    transpose_k<<<tg, tb, 0, stream>>>(Ws1, w1st);
    transpose_k<<<tg, tb, 0, stream>>>(Wm1, w1mt);
    transpose_k<<<tg, tb, 0, stream>>>(Ws2, w2st);
    transpose_k<<<tg, tb, 0, stream>>>(Wm2, w2mt);

    // prep: new_nodes = nodes with x scattered in; passthrough outputs
    size_t n4 = nNODES / 4;
    copy_nodes_k<<<(int)((n4 + 255) / 256), 256, 0, stream>>>((const float4*)nodes,
                                                              (float4*)o_nodes, n4);
    size_t nx4 = nMX / 4;
    scatter_x_k<<<(int)((nx4 + 255) / 256), 256, 0, stream>>>((const float4*)x, T,
                                                              (float4*)o_nodes);
    size_t nmisc = nE2 + nW + BB;
    misc_out_k<<<(int)((nmisc + 255) / 256), 256, 0, stream>>>(edges, weights, T, taus,
                                                               o_edges, o_w, o_t);

    // layer 1
    hipMemsetAsync(agg, 0, nNODES * sizeof(float), stream);
    agg_k<<<(BB * EE) / 8, 256, 0, stream>>>(o_nodes, edges, weights, agg);
    gemm_full_k<<<(BB * NN) / 16, 256, 0, stream>>>(o_nodes, agg, w1st, w1mt, h1);

    // layer 2 (only the 16 output rows per batch are ever gathered)
    hipMemsetAsync(agg, 0, nNODES * sizeof(float), stream);
    agg_k<<<(BB * EE) / 8, 256, 0, stream>>>(h1, edges, weights, agg);
    gemm_out_k<<<BB, 256, 0, stream>>>(h1, agg, w2st, w2mt, T, o_mx);
}